// FLinemodel_37185826849029
// MI455X (gfx1250) — compile-verified
//
#include <hip/hip_runtime.h>
#include <hip/hip_bf16.h>

// Problem constants (match reference)
#define Bq 32
#define Sq 2048
#define Hq 256
#define Dq 4
#define Cq 10

typedef float v2f __attribute__((ext_vector_type(2)));
typedef float v8f __attribute__((ext_vector_type(8)));

// D = A(16x4) * B(4x16) + C, f32, wave32.
__device__ __forceinline__ v8f wmma_f32_16x16x4(v2f a, v2f b, v8f c) {
    return __builtin_amdgcn_wmma_f32_16x16x4_f32(
        /*neg_a=*/false, a, /*neg_b=*/false, b,
        /*c_mod=*/(short)0, c, /*reuse_a=*/false, /*reuse_b=*/false);
}

// --- CDNA5 async global->LDS path (ASYNCcnt-tracked, bypasses VGPRs) -------
// Low 32 bits of a generic pointer into LDS == LDS byte offset (ISA 10.2
// aperture mapping).
__device__ __forceinline__ unsigned lds_off(const void* p) {
    return (unsigned)(uintptr_t)p;
}
// 16 bytes per lane, 512B per wave32 instruction.
__device__ __forceinline__ void async_copy_b128(unsigned lds_byte_off,
                                                const float* gptr) {
    asm volatile("global_load_async_to_lds_b128 %0, %1, off"
                 :: "v"(lds_byte_off), "v"((unsigned long long)(uintptr_t)gptr)
                 : "memory");
}
__device__ __forceinline__ void wait_async_all() {
    asm volatile("s_wait_asynccnt 0x0" ::: "memory");
}
__device__ __forceinline__ void wait_async_16() {   // <=16 outstanding
    asm volatile("s_wait_asynccnt 0x10" ::: "memory");
}

// ---------------------------------------------------------------------------
// Kernel 0: zero the output accumulator (out is accumulated with atomics).
// ---------------------------------------------------------------------------
__global__ void k_zero(float* __restrict__ out, int n) {
    int i = blockIdx.x * blockDim.x + threadIdx.x;
    if (i < n) out[i] = 0.0f;
}

// ---------------------------------------------------------------------------
// Kernel 1: pack wq|wk|wv (each [256,4]) into Wpack [256,16] (cols 12..15 = 0)
// so the QKV projection is exactly one WMMA tile wide.
// ---------------------------------------------------------------------------
__global__ void k_pack(const float* __restrict__ wq, const float* __restrict__ wk,
                       const float* __restrict__ wv, float* __restrict__ wp) {
    int i = blockIdx.x * blockDim.x + threadIdx.x;
    if (i >= Hq * 16) return;
    int h = i >> 4, j = i & 15;
    float v = 0.0f;
    if (j < 4)       v = wq[h * 4 + j];
    else if (j < 8)  v = wk[h * 4 + (j - 4)];
    else if (j < 12) v = wv[h * 4 + (j - 8)];
    wp[i] = v;
}

// ---------------------------------------------------------------------------
// Kernel 2: QKV = x[B*S,256] @ Wpack[256,16], one wave per 16-row tile.
// x tile + weights staged with GLOBAL_LOAD_ASYNC_TO_LDS_B128, then 64 chained
// V_WMMA_F32_16X16X4_F32. Also writes transposed Kt[b][d][t] so the next GEMM
// gets a contiguous A-operand.
// ---------------------------------------------------------------------------
__global__ void __launch_bounds__(32)
k_qkv(const float* __restrict__ x, const float* __restrict__ wp,
      float* __restrict__ qkv, float* __restrict__ Kt) {
    __shared__ alignas(16) float xt[16 * Hq];   // 16 KB: 16 contiguous x rows
    __shared__ alignas(16) float wl[Hq * 16];   // 16 KB: packed weights
    const int lane = threadIdx.x;
    const int r0 = blockIdx.x * 16;             // row tile over B*S rows

    const unsigned xt0 = lds_off(xt);
    const unsigned wl0 = lds_off(wl);
    const float* xg = x + (size_t)r0 * Hq;      // tile is contiguous (4096 f32)
    for (int j = 0; j < 32; ++j) {
        const int q = j * 32 + lane;            // 16-byte chunk index, 0..1023
        async_copy_b128(xt0 + (unsigned)q * 16, xg + q * 4);
        async_copy_b128(wl0 + (unsigned)q * 16, wp + q * 4);
    }
    wait_async_all();
    __syncthreads();

    const int m = lane & 15, half = lane >> 4;
    v8f c = {};
    for (int kc = 0; kc < Hq / 4; ++kc) {
        const int k0 = kc * 4 + 2 * half;
        v2f a, b;
        a.x = xt[m * Hq + k0];      a.y = xt[m * Hq + k0 + 1];   // A[m, k]
        b.x = wl[k0 * 16 + m];      b.y = wl[(k0 + 1) * 16 + m]; // B[k, n=m]
        c = wmma_f32_16x16x4(a, b, c);
    }
    // C/D layout: c[i] = D[row = i + 8*half, col = m]
    for (int i = 0; i < 8; ++i) {
        const int row = r0 + i + 8 * half;
        qkv[(size_t)row * 16 + m] = c[i];
        if (m >= 4 && m < 8) {                 // K columns -> transposed copy
            const int b = row >> 11;           // Sq == 2048
            const int t = row & (Sq - 1);
            const int d = m - 4;
            Kt[((size_t)b * 4 + d) * Sq + t] = c[i];
        }
    }
}

// ---------------------------------------------------------------------------
// Kernel 3: M[b] = K[b]^T @ w_at ([4,2048] per batch), M-dim padded to 16 by
// stacking 4 batches (rows r = bb*4 + d). Double-buffered async global->LDS
// pipeline: while 16 WMMAs consume chunk i (64 t-rows), chunk i+1 streams in.
// Single wave per block => s_wait_asynccnt alone gives LDS visibility.
// ---------------------------------------------------------------------------
__global__ void __launch_bounds__(32)
k_matw(const float* __restrict__ Kt, const float* __restrict__ w_at,
       float* __restrict__ Mws) {
    __shared__ alignas(16) float abuf[2][16 * 64];  // A chunk: [r][trel]
    __shared__ alignas(16) float bbuf[2][64 * 16];  // B chunk: [trel][n]
    const int lane = threadIdx.x;
    const int g  = blockIdx.x;        // batch group of 4
    const int u0 = blockIdx.y * 16;   // output column tile
    const int m = lane & 15, half = lane >> 4;

    auto stage = [&](int cc, int pb) {
        const int t0 = cc * 64;
        const unsigned a0 = lds_off(abuf[pb]);
        const unsigned b0 = lds_off(bbuf[pb]);
        for (int j = 0; j < 8; ++j) {
            const int q = j * 32 + lane;          // 0..255 16-byte chunks
            {   // A: 16 rows x 64 floats, rows contiguous in Kt
                const int r = q >> 4, seg = q & 15;
                async_copy_b128(a0 + (unsigned)(r * 64 + seg * 4) * 4,
                                Kt + ((size_t)g * 16 + r) * Sq + t0 + seg * 4);
            }
            {   // B: 64 rows x 16 floats of w_at
                const int tr = q >> 2, seg = q & 3;
                async_copy_b128(b0 + (unsigned)(tr * 16 + seg * 4) * 4,
                                w_at + (size_t)(t0 + tr) * Sq + u0 + seg * 4);
            }
        }
    };

    stage(0, 0);
    v8f c = {};
    for (int cc = 0; cc < Sq / 64; ++cc) {
        const int pb = cc & 1;
        if (cc + 1 < Sq / 64) {
            stage(cc + 1, pb ^ 1);   // 16 asyncs in flight for next chunk
            wait_async_16();         // in-order completion => chunk cc landed
        } else {
            wait_async_all();
        }
        for (int kk = 0; kk < 16; ++kk) {
            const int tr = kk * 4 + 2 * half;
            v2f a, bf;
            a.x  = abuf[pb][m * 64 + tr];
            a.y  = abuf[pb][m * 64 + tr + 1];
            bf.x = bbuf[pb][tr * 16 + m];
            bf.y = bbuf[pb][(tr + 1) * 16 + m];
            c = wmma_f32_16x16x4(a, bf, c);
        }
    }
    // row r = i + 8*half; Mws row index (g*16 + r) == (b*4 + d)
    for (int i = 0; i < 8; ++i) {
        const int r = i + 8 * half;
        Mws[((size_t)g * 16 + r) * Sq + u0 + m] = c[i];
    }
}

// ---------------------------------------------------------------------------
// Kernel 4: fused  scores = Q@M + b_at  -> softmax (rows live in LDS)
//                  y = P@V ; p = softmax(y@w_cls + b_cls) ; mean over s.
// Grid = (B, S/16), block = 128 (4 waves). LDS: 16x2048 f32 score tile
// (128 KB) + reductions => 2 blocks per WGP (320 KB LDS).
// ---------------------------------------------------------------------------
__global__ void __launch_bounds__(128)
k_attn(const float* __restrict__ qkv, const float* __restrict__ Mws,
       const float* __restrict__ b_at, const float* __restrict__ w_cls,
       const float* __restrict__ b_cls, float* __restrict__ out) {
    extern __shared__ float smem[];
    float* sc     = smem;                  // [16][2048] score tile
    float* part   = sc + 16 * Sq;          // [16][8] partial reductions
    float* rowmax = part + 16 * 8;         // [16]
    float* rowsum = rowmax + 16;           // [16]
    float* ypart  = rowsum + 16;           // [4][16][4] per-wave y partials
    float* plog   = ypart + 4 * 16 * 4;    // [16][10] class probabilities

    const int tid  = threadIdx.x;
    const int lane = tid & 31;
    const int w    = tid >> 5;             // wave id 0..3
    const int bi   = blockIdx.x;           // batch
    const int s0   = blockIdx.y * 16;      // row tile
    const int m = lane & 15, half = lane >> 4;

    // Q fragment for this 16-row tile (reused for all 128 u-tiles)
    v2f qa;
    qa.x = qkv[((size_t)bi * Sq + s0 + m) * 16 + 2 * half];
    qa.y = qkv[((size_t)bi * Sq + s0 + m) * 16 + 2 * half + 1];

    // --- raw scores: one WMMA per 16x16 tile, waves split the 128 u-tiles ---
    for (int ut = w; ut < Sq / 16; ut += 4) {
        const int u0 = ut * 16;
        v2f mb;
        mb.x = Mws[((size_t)bi * 4 + 2 * half) * Sq + u0 + m];
        mb.y = Mws[((size_t)bi * 4 + 2 * half + 1) * Sq + u0 + m];
        v8f c = {};
        c = wmma_f32_16x16x4(qa, mb, c);
        const float bias = b_at[u0 + m];
        for (int i = 0; i < 8; ++i)
            sc[(i + 8 * half) * Sq + u0 + m] = c[i] + bias;
    }
    __syncthreads();

    // --- row max (8 threads per row) ---
    {
        const int r = tid >> 3, o = tid & 7;
        float mx = -3.4e38f;
        for (int u = o; u < Sq; u += 8) mx = fmaxf(mx, sc[r * Sq + u]);
        part[r * 8 + o] = mx;
    }
    __syncthreads();
    if (tid < 16) {
        float mx = part[tid * 8];
        for (int o = 1; o < 8; ++o) mx = fmaxf(mx, part[tid * 8 + o]);
        rowmax[tid] = mx;
    }
    __syncthreads();

    // --- exp in place + row sum ---
    {
        const int r = tid >> 3, o = tid & 7;
        const float mx = rowmax[r];
        float sm = 0.0f;
        for (int u = o; u < Sq; u += 8) {
            const float e = __expf(sc[r * Sq + u] - mx);
            sc[r * Sq + u] = e;
            sm += e;
        }
        part[r * 8 + o] = sm;
    }
    __syncthreads();
    if (tid < 16) {
        float sm = 0.0f;
        for (int o = 0; o < 8; ++o) sm += part[tid * 8 + o];
        rowsum[tid] = sm;
    }
    __syncthreads();

    // --- y = P[16,2048] @ V[2048,4] via WMMA K-chunks of 4 (V padded to 16) ---
    v8f cy = {};
    for (int uc = w; uc < Sq / 4; uc += 4) {
        const int t0 = uc * 4 + 2 * half;
        v2f a, bf;
        a.x  = sc[m * Sq + t0];
        a.y  = sc[m * Sq + t0 + 1];
        bf.x = (m < 4) ? qkv[((size_t)bi * Sq + t0) * 16 + 8 + m] : 0.0f;
        bf.y = (m < 4) ? qkv[((size_t)bi * Sq + t0 + 1) * 16 + 8 + m] : 0.0f;
        cy = wmma_f32_16x16x4(a, bf, cy);
    }
    if (m < 4)
        for (int i = 0; i < 8; ++i)
            ypart[(w * 16 + i + 8 * half) * 4 + m] = cy[i];
    __syncthreads();

    // --- reduce y across waves, normalize, classifier softmax per row ---
    if (tid < 16) {
        const int r = tid;
        float y0 = 0, y1 = 0, y2 = 0, y3 = 0;
        for (int wv = 0; wv < 4; ++wv) {
            y0 += ypart[(wv * 16 + r) * 4 + 0];
            y1 += ypart[(wv * 16 + r) * 4 + 1];
            y2 += ypart[(wv * 16 + r) * 4 + 2];
            y3 += ypart[(wv * 16 + r) * 4 + 3];
        }
        const float inv = 1.0f / rowsum[r];
        y0 *= inv; y1 *= inv; y2 *= inv; y3 *= inv;
        float lg[Cq];
        float mx = -3.4e38f;
        for (int cc = 0; cc < Cq; ++cc) {
            float l = b_cls[cc] + y0 * w_cls[0 * Cq + cc] + y1 * w_cls[1 * Cq + cc]
                                + y2 * w_cls[2 * Cq + cc] + y3 * w_cls[3 * Cq + cc];
            lg[cc] = l;
            mx = fmaxf(mx, l);
        }
        float sm = 0.0f;
        for (int cc = 0; cc < Cq; ++cc) { lg[cc] = __expf(lg[cc] - mx); sm += lg[cc]; }
        const float isv = 1.0f / sm;
        for (int cc = 0; cc < Cq; ++cc) plog[r * Cq + cc] = lg[cc] * isv;
    }
    __syncthreads();

    // --- per-block class sums -> one atomicAdd per class (mean over S) ---
    if (tid < Cq) {
        float acc = 0.0f;
        for (int r = 0; r < 16; ++r) acc += plog[r * Cq + tid];
        atomicAdd(&out[bi * Cq + tid], acc * (1.0f / (float)Sq));
    }
}

// ---------------------------------------------------------------------------
extern "C" void kernel_launch(void* const* d_in, const int* in_sizes, int n_in,
                              void* d_out, int out_size, void* d_ws, size_t ws_size,
                              hipStream_t stream) {
    const float* x     = (const float*)d_in[0];
    const float* wq    = (const float*)d_in[1];
    const float* wk    = (const float*)d_in[2];
    const float* wv    = (const float*)d_in[3];
    const float* w_at  = (const float*)d_in[4];
    const float* b_at  = (const float*)d_in[5];
    const float* w_cls = (const float*)d_in[6];
    const float* b_cls = (const float*)d_in[7];
    float* out = (float*)d_out;

    // workspace layout (floats): Wpack | QKV | M | Kt   (~6.3 MB total)
    float* wp  = (float*)d_ws;                              // Hq*16
    float* qkv = wp + (size_t)Hq * 16;                      // Bq*Sq*16
    float* Mws = qkv + (size_t)Bq * Sq * 16;                // Bq*4*Sq
    float* Kt  = Mws + (size_t)Bq * 4 * Sq;                 // Bq*4*Sq

    k_zero<<<(Bq * Cq + 255) / 256, 256, 0, stream>>>(out, Bq * Cq);
    k_pack<<<(Hq * 16 + 255) / 256, 256, 0, stream>>>(wq, wk, wv, wp);
    k_qkv<<<(Bq * Sq) / 16, 32, 0, stream>>>(x, wp, qkv, Kt);

    dim3 gm(Bq / 4, Sq / 16);
    k_matw<<<gm, 32, 0, stream>>>(Kt, w_at, Mws);

    dim3 ga(Bq, Sq / 16);
    const size_t shmem =
        (size_t)(16 * Sq + 16 * 8 + 16 + 16 + 4 * 16 * 4 + 16 * Cq) * sizeof(float);
    k_attn<<<ga, 128, shmem, stream>>>(qkv, Mws, b_at, w_cls, b_cls, out);
}